// QuantumLayerVectorized_32787780337762
// MI455X (gfx1250) — compile-verified
//
#include <hip/hip_runtime.h>
#include <hip/hip_bf16.h>

// Quantum circuit simulator for 12 qubits, depth-4 RX+CNOT-chain layers, batch 8192.
// One workgroup (128 threads = 4 waves) per batch element; state lives in LDS
// (ping-pong re/im planes). RX layers are applied as three 16x16 complex matmuls
// (4-qubit groups) using V_WMMA_F32_16X16X4_F32. The CNOT chain composes to the
// bit permutation y -> y ^ (y>>1), folded into the next read pass.

typedef float v2f __attribute__((ext_vector_type(2)));
typedef float v8f __attribute__((ext_vector_type(8)));

#define NQ      12
#define DIM     4096
#define DEPTH   4
#define THREADS 128

static __device__ __forceinline__ v8f wmma4(v2f a, v2f b, v8f c) {
  // D(16x16) += A(16x4) @ B(4x16), all f32. 8-arg form per CDNA5 bridge doc.
  return __builtin_amdgcn_wmma_f32_16x16x4_f32(false, a, false, b, (short)0, c,
                                               false, false);
}

// Entry (r,c) of the 16x16 group matrix  U_g = RX(w[4g+0]) (x) ... (x) RX(w[4g+3]).
// RX = [[cos, -i sin], [-i sin, cos]] -> entry = (-i)^popcount(r^c) * prod(cos/sin).
// Bit 3 of r/c corresponds to the first wire of the group (MSB side).
static __device__ __forceinline__ void uentry(int r, int c,
                                              const float* lwc, const float* lws,
                                              int g, float& re, float& im) {
  const int x = r ^ c;
  float v = 1.0f;
#pragma unroll
  for (int bp = 3; bp >= 0; --bp) {
    const int w = 4 * g + (3 - bp);
    v *= ((x >> bp) & 1) ? lws[w] : lwc[w];
  }
  const int m = __popc(x) & 3;  // phase (-i)^m
  re = (m == 0) ? v : ((m == 2) ? -v : 0.0f);
  im = (m == 1) ? -v : ((m == 3) ? v : 0.0f);
}

// Build the 4 K-chunk fragments of U (re, im, -im). Lane layout works both as
// A-operand (A[M][K] = U[M][K], M = lane&15) and as B-operand (B[K][N] = U[N][K],
// N = lane&15, i.e. U transposed — exactly what step C needs).
static __device__ __forceinline__ void ufrag(const float* lwc, const float* lws,
                                             int g, int lane,
                                             v2f ur[4], v2f ui[4], v2f uin[4]) {
  const int r  = lane & 15;
  const int hk = (lane >> 4) << 1;  // 0 for lanes 0-15, 2 for lanes 16-31
#pragma unroll
  for (int k = 0; k < 4; ++k) {
    float r0, i0, r1, i1;
    uentry(r, 4 * k + hk,     lwc, lws, g, r0, i0);
    uentry(r, 4 * k + hk + 1, lwc, lws, g, r1, i1);
    ur[k].x  = r0;  ur[k].y  = r1;
    ui[k].x  = i0;  ui[k].y  = i1;
    uin[k].x = -i0; uin[k].y = -i1;
  }
}

// ---- Step C: contract LSB group (wires 8-11).  Psi'[ab, c] = sum_c' UC[c,c'] Psi[ab, c'].
// Computed as D = S @ UC^T  (state rows contiguous in memory are the A operand).
// Optionally reads through the CNOT-chain permutation x = y ^ (y>>1).
static __device__ __forceinline__ void step_c(const float* sR, const float* sI,
                                              float* dR, float* dI, bool perm,
                                              const float* lwc, const float* lws,
                                              int lane, int wave) {
  v2f ur[4], ui[4], uin[4];
  ufrag(lwc, lws, 2, lane, ur, ui, uin);
  const int M      = lane & 15;
  const int hk     = (lane >> 4) << 1;
  const int rowsel = 8 * (lane >> 4);
#pragma unroll
  for (int i = 0; i < 4; ++i) {
    const int t = 4 * wave + i;
    const int R = 16 * t + M;
    v2f ar[4], ai[4];
#pragma unroll
    for (int k = 0; k < 4; ++k) {
      const int y0 = R * 16 + 4 * k + hk;
      const int y1 = y0 + 1;
      const int x0 = perm ? (y0 ^ (y0 >> 1)) : y0;
      const int x1 = perm ? (y1 ^ (y1 >> 1)) : y1;
      ar[k].x = sR[x0]; ar[k].y = sR[x1];
      ai[k].x = sI[x0]; ai[k].y = sI[x1];
    }
    v8f dr = {0, 0, 0, 0, 0, 0, 0, 0};
    v8f di = {0, 0, 0, 0, 0, 0, 0, 0};
#pragma unroll
    for (int k = 0; k < 4; ++k) {
      dr = wmma4(ar[k], ur[k], dr);   // Sre @ Ure
      dr = wmma4(ai[k], uin[k], dr);  // - Sim @ Uim
      di = wmma4(ar[k], ui[k], di);   // Sre @ Uim
      di = wmma4(ai[k], ur[k], di);   // Sim @ Ure
    }
#pragma unroll
    for (int v = 0; v < 8; ++v) {
      const int addr = (16 * t + rowsel + v) * 16 + M;
      dR[addr] = dr[v];
      dI[addr] = di[v];
    }
  }
  __syncthreads();
}

// ---- Step B: contract middle group (wires 4-7). Per a-block: D = UB @ S_a.
static __device__ __forceinline__ void step_b(const float* sR, const float* sI,
                                              float* dR, float* dI,
                                              const float* lwc, const float* lws,
                                              int lane, int wave) {
  v2f ur[4], ui[4], uin[4];
  ufrag(lwc, lws, 1, lane, ur, ui, uin);
  const int N      = lane & 15;
  const int hk     = (lane >> 4) << 1;
  const int rowsel = 8 * (lane >> 4);
#pragma unroll
  for (int i = 0; i < 4; ++i) {
    const int base = (4 * wave + i) * 256;
    v2f br[4], bi[4];
#pragma unroll
    for (int k = 0; k < 4; ++k) {
      const int a0 = base + (4 * k + hk) * 16 + N;
      br[k].x = sR[a0]; br[k].y = sR[a0 + 16];
      bi[k].x = sI[a0]; bi[k].y = sI[a0 + 16];
    }
    v8f dr = {0, 0, 0, 0, 0, 0, 0, 0};
    v8f di = {0, 0, 0, 0, 0, 0, 0, 0};
#pragma unroll
    for (int k = 0; k < 4; ++k) {
      dr = wmma4(ur[k], br[k], dr);   // Ure @ Sre
      dr = wmma4(uin[k], bi[k], dr);  // - Uim @ Sim
      di = wmma4(ur[k], bi[k], di);   // Ure @ Sim
      di = wmma4(ui[k], br[k], di);   // Uim @ Sre
    }
#pragma unroll
    for (int v = 0; v < 8; ++v) {
      const int addr = base + (rowsel + v) * 16 + N;
      dR[addr] = dr[v];
      dI[addr] = di[v];
    }
  }
  __syncthreads();
}

// ---- Step A: contract MSB group (wires 0-3). Per column slab: D = UA @ S.
static __device__ __forceinline__ void step_a(const float* sR, const float* sI,
                                              float* dR, float* dI,
                                              const float* lwc, const float* lws,
                                              int lane, int wave) {
  v2f ur[4], ui[4], uin[4];
  ufrag(lwc, lws, 0, lane, ur, ui, uin);
  const int N      = lane & 15;
  const int hk     = (lane >> 4) << 1;
  const int rowsel = 8 * (lane >> 4);
#pragma unroll
  for (int i = 0; i < 4; ++i) {
    const int cb = (4 * wave + i) * 16;
    v2f br[4], bi[4];
#pragma unroll
    for (int k = 0; k < 4; ++k) {
      const int a0 = (4 * k + hk) * 256 + cb + N;
      br[k].x = sR[a0]; br[k].y = sR[a0 + 256];
      bi[k].x = sI[a0]; bi[k].y = sI[a0 + 256];
    }
    v8f dr = {0, 0, 0, 0, 0, 0, 0, 0};
    v8f di = {0, 0, 0, 0, 0, 0, 0, 0};
#pragma unroll
    for (int k = 0; k < 4; ++k) {
      dr = wmma4(ur[k], br[k], dr);
      dr = wmma4(uin[k], bi[k], dr);
      di = wmma4(ur[k], bi[k], di);
      di = wmma4(ui[k], br[k], di);
    }
#pragma unroll
    for (int v = 0; v < 8; ++v) {
      const int addr = (rowsel + v) * 256 + cb + N;
      dR[addr] = dr[v];
      dI[addr] = di[v];
    }
  }
  __syncthreads();
}

__global__ void __launch_bounds__(THREADS)
qsim_kernel(const float* __restrict__ inp, const float* __restrict__ wts,
            float* __restrict__ out) {
  __shared__ float S0R[DIM], S0I[DIM], S1R[DIM], S1I[DIM];  // 64 KB ping-pong state
  __shared__ float wc[DEPTH * NQ], ws[DEPTH * NQ];          // cos/sin of weights/2
  __shared__ float tc[NQ], ts[NQ];                          // cos/sin of inputs/2
  __shared__ float red[NQ];

  const int tid  = threadIdx.x;
  const int lane = tid & 31;
  const int wave = tid >> 5;
  const int b    = blockIdx.x;

  if (tid < NQ) {
    const float h = 0.5f * inp[b * NQ + tid];
    tc[tid] = cosf(h);
    ts[tid] = sinf(h);
    red[tid] = 0.0f;
  }
  if (tid < DEPTH * NQ) {
    const float h = 0.5f * wts[tid];
    wc[tid] = cosf(h);
    ws[tid] = sinf(h);
  }
  __syncthreads();

  // RY-encoded product state (purely real): psi[x] = prod_q (bit ? sin : cos).
  for (int i = tid; i < DIM; i += THREADS) {
    float v = 1.0f;
#pragma unroll
    for (int q = 0; q < NQ; ++q)
      v *= ((i >> (11 - q)) & 1) ? ts[q] : tc[q];
    S0R[i] = v;
    S0I[i] = 0.0f;
  }
  __syncthreads();

  // Two layers per iteration keeps buffer pointers compile-time constant.
  // Layer even: buf0 -C-> buf1 -B-> buf0 -A-> buf1 ; layer odd: buf1 -> ... -> buf0.
  // Each layer's trailing CNOT chain is folded into the NEXT step_c read (perm).
#pragma unroll 1
  for (int it = 0; it < 2; ++it) {
    const int d0 = 2 * it, d1 = 2 * it + 1;
    step_c(S0R, S0I, S1R, S1I, it > 0, &wc[d0 * NQ], &ws[d0 * NQ], lane, wave);
    step_b(S1R, S1I, S0R, S0I,         &wc[d0 * NQ], &ws[d0 * NQ], lane, wave);
    step_a(S0R, S0I, S1R, S1I,         &wc[d0 * NQ], &ws[d0 * NQ], lane, wave);

    step_c(S1R, S1I, S0R, S0I, true,   &wc[d1 * NQ], &ws[d1 * NQ], lane, wave);
    step_b(S0R, S0I, S1R, S1I,         &wc[d1 * NQ], &ws[d1 * NQ], lane, wave);
    step_a(S1R, S1I, S0R, S0I,         &wc[d1 * NQ], &ws[d1 * NQ], lane, wave);
  }

  // <Z_q> with the final CNOT-chain permutation folded into the read index.
  float acc[NQ];
#pragma unroll
  for (int q = 0; q < NQ; ++q) acc[q] = 0.0f;
  for (int y = tid; y < DIM; y += THREADS) {
    const int x = y ^ (y >> 1);
    const float re = S0R[x], im = S0I[x];
    const float p = re * re + im * im;
#pragma unroll
    for (int q = 0; q < NQ; ++q)
      acc[q] += ((y >> (11 - q)) & 1) ? -p : p;
  }
#pragma unroll
  for (int q = 0; q < NQ; ++q) {
#pragma unroll
    for (int o = 16; o > 0; o >>= 1) acc[q] += __shfl_xor(acc[q], o, 32);
  }
  if (lane == 0) {
#pragma unroll
    for (int q = 0; q < NQ; ++q) atomicAdd(&red[q], acc[q]);
  }
  __syncthreads();
  if (tid < NQ) out[b * NQ + tid] = red[tid];
}

extern "C" void kernel_launch(void* const* d_in, const int* in_sizes, int n_in,
                              void* d_out, int out_size, void* d_ws, size_t ws_size,
                              hipStream_t stream) {
  const float* inp = (const float*)d_in[0];  // [BATCH, 12] f32
  const float* wts = (const float*)d_in[1];  // [48] f32
  float* out = (float*)d_out;                // [BATCH, 12] f32
  const int batch = in_sizes[0] / NQ;
  qsim_kernel<<<dim3(batch), dim3(THREADS), 0, stream>>>(inp, wts, out);
}